// LatentSDE_36146444763912
// MI455X (gfx1250) — compile-verified
//
#include <hip/hip_runtime.h>

typedef __attribute__((ext_vector_type(2))) float v2f;
typedef __attribute__((ext_vector_type(8))) float v8f;

namespace {
constexpr int kB = 4096;
constexpr int kD = 16;
constexpr int kT = 1000;
constexpr int kH = 48;
constexpr int kWavesPerBlock = 4;
constexpr int kRowsPerWave = 16;
constexpr int kThreads = kWavesPerBlock * 32;          // 128
constexpr int kBlocks = kB / (kWavesPerBlock * kRowsPerWave); // 64
}

__device__ __forceinline__ v8f wmma4(v2f a, v2f b, v8f c) {
#if defined(__gfx1250__)
  // D = A(16x4 f32) * B(4x16 f32) + C(16x16 f32)
  return __builtin_amdgcn_wmma_f32_16x16x4_f32(
      /*neg_a=*/false, a, /*neg_b=*/false, b,
      /*c_mod=*/(short)0, c, /*reuse_a=*/false, /*reuse_b=*/false);
#else
  (void)a; (void)b;
  return c;
#endif
}

__device__ __forceinline__ float fast_tanh(float x) {
  // tanh(x) = (e^{2x} - 1) / (e^{2x} + 1), via v_exp_f32 + v_rcp_f32
  float e = __expf(2.0f * x);
  return (e - 1.0f) * __builtin_amdgcn_rcpf(e + 1.0f);
}

#if defined(__gfx1250__)
#define WAVE_LDS_SYNC() asm volatile("s_wait_dscnt 0x0" ::: "memory")
#else
#define WAVE_LDS_SYNC() __threadfence_block()
#endif

__global__ __launch_bounds__(kThreads) void latent_sde_kernel(
    const float* __restrict__ z0, const float* __restrict__ ts,
    const float* __restrict__ W1, const float* __restrict__ b1,
    const float* __restrict__ W2, const float* __restrict__ b2,
    const float* __restrict__ W3, const float* __restrict__ b3,
    const float* __restrict__ log_std, const float* __restrict__ noise,
    float* __restrict__ out) {
  const int lane = threadIdx.x & 31;
  const int wave = threadIdx.x >> 5;
  const int rowBase = (blockIdx.x * kWavesPerBlock + wave) * kRowsPerWave;

  const int nl   = lane & 15;   // A: row m; B/C: column n
  const int hi   = lane >> 4;   // low/high half-wave
  const int koff = hi * 2;      // A/B fragment K offset
  const int mOff = hi * 8;      // C/D row offset

  // Per-wave private LDS staging buffers (strides 21/49 -> no 64-bank conflicts)
  __shared__ float xbufAll[kWavesPerBlock][16][21];
  __shared__ float hbufAll[kWavesPerBlock][16][49];
  float (*xbuf)[21] = xbufAll[wave];
  float (*hbuf)[49] = hbufAll[wave];

  // Zero the K-padding columns (K padded 17 -> 20) once; never overwritten.
  if (hi == 0) {
    xbuf[nl][17] = 0.0f;
    xbuf[nl][18] = 0.0f;
    xbuf[nl][19] = 0.0f;
  }

  // ---- Pack weights into register-resident B-fragments (loaded once) ----
  // B frag for k-step k, n-tile j: rows 4k+koff, 4k+koff+1; column j*16+nl.
  v2f w1f[5][3];
#pragma unroll
  for (int k = 0; k < 5; ++k) {
    const int r0 = 4 * k + koff;
#pragma unroll
    for (int j = 0; j < 3; ++j) {
      const int c = j * 16 + nl;
      w1f[k][j].x = (r0 < 17) ? W1[r0 * kH + c] : 0.0f;
      w1f[k][j].y = (r0 + 1 < 17) ? W1[(r0 + 1) * kH + c] : 0.0f;
    }
  }
  v2f w2f[12][3];
#pragma unroll
  for (int k = 0; k < 12; ++k) {
    const int r0 = 4 * k + koff;
#pragma unroll
    for (int j = 0; j < 3; ++j) {
      const int c = j * 16 + nl;
      w2f[k][j].x = W2[r0 * kH + c];
      w2f[k][j].y = W2[(r0 + 1) * kH + c];
    }
  }
  v2f w3f[12];
#pragma unroll
  for (int k = 0; k < 12; ++k) {
    const int r0 = 4 * k + koff;
    w3f[k].x = W3[r0 * kD + nl];
    w3f[k].y = W3[(r0 + 1) * kD + nl];
  }

  // Per-lane biases (C layout: lane holds column nl of each tile)
  float b1l[3], b2l[3];
#pragma unroll
  for (int j = 0; j < 3; ++j) {
    b1l[j] = b1[j * 16 + nl];
    b2l[j] = b2[j * 16 + nl];
  }
  const float b3l  = b3[nl];
  const float stdl = __expf(log_std[nl]);

  // ---- Initial state z0 in C layout: zc[v] = z[rowBase + mOff + v][nl] ----
  float zc[8];
#pragma unroll
  for (int v = 0; v < 8; ++v)
    zc[v] = z0[(rowBase + mOff + v) * kD + nl];

  // out[0] = z0
#pragma unroll
  for (int v = 0; v < 8; ++v)
    out[(size_t)(rowBase + mOff + v) * kD + nl] = zc[v];

  for (int step = 0; step < kT - 1; ++step) {
    const float tcur = ts[step];
    const float dt   = ts[step + 1] - tcur;
    const float sdw  = stdl * sqrtf(dt);

    // ---- Stage x = [t, z] into LDS (row-major 16x20, padded K) ----
    if (hi == 0) xbuf[nl][0] = tcur;
#pragma unroll
    for (int v = 0; v < 8; ++v)
      xbuf[mOff + v][1 + nl] = zc[v];
    WAVE_LDS_SYNC();

    // ---- Layer 1: [16,20] @ [20,48] ----
    v8f acc0 = {}, acc1 = {}, acc2 = {};
#pragma unroll
    for (int k = 0; k < 5; ++k) {
      v2f a;
      a.x = xbuf[nl][4 * k + koff];
      a.y = xbuf[nl][4 * k + koff + 1];
      acc0 = wmma4(a, w1f[k][0], acc0);
      acc1 = wmma4(a, w1f[k][1], acc1);
      acc2 = wmma4(a, w1f[k][2], acc2);
    }
#pragma unroll
    for (int v = 0; v < 8; ++v) {
      hbuf[mOff + v][ 0 + nl] = fast_tanh(acc0[v] + b1l[0]);
      hbuf[mOff + v][16 + nl] = fast_tanh(acc1[v] + b1l[1]);
      hbuf[mOff + v][32 + nl] = fast_tanh(acc2[v] + b1l[2]);
    }
    WAVE_LDS_SYNC();

    // ---- Layer 2: [16,48] @ [48,48] ----
    v8f bcc0 = {}, bcc1 = {}, bcc2 = {};
#pragma unroll
    for (int k = 0; k < 12; ++k) {
      v2f a;
      a.x = hbuf[nl][4 * k + koff];
      a.y = hbuf[nl][4 * k + koff + 1];
      bcc0 = wmma4(a, w2f[k][0], bcc0);
      bcc1 = wmma4(a, w2f[k][1], bcc1);
      bcc2 = wmma4(a, w2f[k][2], bcc2);
    }
#pragma unroll
    for (int v = 0; v < 8; ++v) {
      hbuf[mOff + v][ 0 + nl] = fast_tanh(bcc0[v] + b2l[0]);
      hbuf[mOff + v][16 + nl] = fast_tanh(bcc1[v] + b2l[1]);
      hbuf[mOff + v][32 + nl] = fast_tanh(bcc2[v] + b2l[2]);
    }
    WAVE_LDS_SYNC();

    // ---- Layer 3: [16,48] @ [48,16] -> drift f in C layout ----
    v8f facc = {};
#pragma unroll
    for (int k = 0; k < 12; ++k) {
      v2f a;
      a.x = hbuf[nl][4 * k + koff];
      a.y = hbuf[nl][4 * k + koff + 1];
      facc = wmma4(a, w3f[k], facc);
    }

    // ---- Euler-Maruyama update + streaming noise in / z out ----
    const float* nzp = noise + (size_t)step * kB * kD;
    float* op = out + (size_t)(step + 1) * kB * kD;
#pragma unroll
    for (int v = 0; v < 8; ++v) {
      const int idx = (rowBase + mOff + v) * kD + nl;
      const float dw = nzp[idx];
      zc[v] = zc[v] + (facc[v] + b3l) * dt + sdw * dw;
      op[idx] = zc[v];
    }
  }
}

extern "C" void kernel_launch(void* const* d_in, const int* in_sizes, int n_in,
                              void* d_out, int out_size, void* d_ws, size_t ws_size,
                              hipStream_t stream) {
  (void)in_sizes; (void)n_in; (void)out_size; (void)d_ws; (void)ws_size;
  const float* z0      = (const float*)d_in[0];
  const float* ts      = (const float*)d_in[1];
  const float* W1      = (const float*)d_in[2];
  const float* b1      = (const float*)d_in[3];
  const float* W2      = (const float*)d_in[4];
  const float* b2      = (const float*)d_in[5];
  const float* W3      = (const float*)d_in[6];
  const float* b3      = (const float*)d_in[7];
  const float* log_std = (const float*)d_in[8];
  const float* noise   = (const float*)d_in[9];
  float* out = (float*)d_out;

  latent_sde_kernel<<<dim3(kBlocks), dim3(kThreads), 0, stream>>>(
      z0, ts, W1, b1, W2, b2, W3, b3, log_std, noise, out);
}